// GraphAttn_50560355008913
// MI455X (gfx1250) — compile-verified
//
#include <hip/hip_runtime.h>
#include <math.h>

#define N_NODES 100000
#define N_EDGES 600000
#define DIM     128
#define LN_EPS  1e-5f
#define LDS_STRIDE 132   // 128 + 4 pad: row stride mod 64 banks = 4 -> conflict-free A frags

typedef __attribute__((ext_vector_type(2))) float v2f;
typedef __attribute__((ext_vector_type(8))) float v8f;

// ---------------------------------------------------------------------------
// Kernel 1: fused LayerNorm + q/k/v projection.
// Block = 256 threads (8 waves) handles one 16-row node tile.
//   Phase 1: LN of 16x128 tile into LDS (16 lanes/row, 8 elems/lane).
//   Phase 2: wave w computes 16x16 output tile (cols w*16..) for q, k AND v,
//            sharing one LDS A-fragment per K-step across 3 WMMAs.
// ---------------------------------------------------------------------------
__global__ __launch_bounds__(256) void lnqkv_kernel(
        const float* __restrict__ x,
        const float* __restrict__ gamma, const float* __restrict__ beta,
        const float* __restrict__ Wq, const float* __restrict__ bq,
        const float* __restrict__ Wk, const float* __restrict__ bk,
        const float* __restrict__ Wv, const float* __restrict__ bv,
        float* __restrict__ q, float* __restrict__ k, float* __restrict__ v) {
    __shared__ float sdelta[16 * LDS_STRIDE];

    const int tid = threadIdx.x;
    const int tr  = blockIdx.x * 16;

    // ---- Phase 1: LayerNorm into LDS -------------------------------------
    {
        const int r  = tid >> 4;        // row in tile, 0..15
        const int c0 = (tid & 15) * 8;  // starting column for this lane
        const float* xr = x + (size_t)(tr + r) * DIM + c0;
        float4 x0 = *(const float4*)xr;
        float4 x1 = *(const float4*)(xr + 4);
        float s  = x0.x + x0.y + x0.z + x0.w + x1.x + x1.y + x1.z + x1.w;
        float sq = x0.x*x0.x + x0.y*x0.y + x0.z*x0.z + x0.w*x0.w
                 + x1.x*x1.x + x1.y*x1.y + x1.z*x1.z + x1.w*x1.w;
        #pragma unroll
        for (int m = 8; m >= 1; m >>= 1) {
            s  += __shfl_xor(s,  m, 16);   // 16 lanes own one row
            sq += __shfl_xor(sq, m, 16);
        }
        const float mu  = s  * (1.0f / DIM);
        const float var = sq * (1.0f / DIM) - mu * mu;
        const float rs  = rsqrtf(var + LN_EPS);

        float4 g0 = *(const float4*)(gamma + c0);
        float4 g1 = *(const float4*)(gamma + c0 + 4);
        float4 b0 = *(const float4*)(beta  + c0);
        float4 b1 = *(const float4*)(beta  + c0 + 4);
        float4 o0, o1;
        o0.x = (x0.x - mu) * rs * g0.x + b0.x;
        o0.y = (x0.y - mu) * rs * g0.y + b0.y;
        o0.z = (x0.z - mu) * rs * g0.z + b0.z;
        o0.w = (x0.w - mu) * rs * g0.w + b0.w;
        o1.x = (x1.x - mu) * rs * g1.x + b1.x;
        o1.y = (x1.y - mu) * rs * g1.y + b1.y;
        o1.z = (x1.z - mu) * rs * g1.z + b1.z;
        o1.w = (x1.w - mu) * rs * g1.w + b1.w;
        *(float4*)&sdelta[r * LDS_STRIDE + c0]     = o0;
        *(float4*)&sdelta[r * LDS_STRIDE + c0 + 4] = o1;
    }
    __syncthreads();

    // ---- Phase 2: 3 WMMA GEMM tiles per wave -----------------------------
    const int lane = tid & 31;
    const int lo   = lane & 15;
    const int hi   = lane >> 4;
    const int tc   = (tid >> 5) * 16;    // wave w -> col tile w

    // A fragment (16x4 f32): lane lo = row M; hi selects K pair {0,1}/{2,3}
    const float* sa = &sdelta[lo * LDS_STRIDE + 2 * hi];
    // B fragments: B[k][n] = W[n][k], lane lo = col N (row-major [out,in] W)
    const float* pq = Wq + (size_t)(tc + lo) * DIM + 2 * hi;
    const float* pk = Wk + (size_t)(tc + lo) * DIM + 2 * hi;
    const float* pv = Wv + (size_t)(tc + lo) * DIM + 2 * hi;

    v8f cq = {}, ck = {}, cv = {};
    #pragma unroll
    for (int s = 0; s < 32; ++s) {
        v2f a   = *(const v2f*)(sa + 4 * s);     // ds_load_b64, shared A
        v2f bbq = *(const v2f*)(pq + 4 * s);
        v2f bbk = *(const v2f*)(pk + 4 * s);
        v2f bbv = *(const v2f*)(pv + 4 * s);
        cq = __builtin_amdgcn_wmma_f32_16x16x4_f32(false, a, false, bbq,
                                                   (short)0, cq, false, false);
        ck = __builtin_amdgcn_wmma_f32_16x16x4_f32(false, a, false, bbk,
                                                   (short)0, ck, false, false);
        cv = __builtin_amdgcn_wmma_f32_16x16x4_f32(false, a, false, bbv,
                                                   (short)0, cv, false, false);
    }

    const int   col = tc + lo;
    const float vbq = bq[col], vbk = bk[col], vbv = bv[col];
    #pragma unroll
    for (int i = 0; i < 8; ++i) {
        const size_t row = (size_t)(tr + i + 8 * hi);   // C layout: VGPR i -> M=i / i+8
        q[row * DIM + col] = (cq[i] + vbq) * 0.25f;     // 1/sqrt(HDIM) folded in
        k[row * DIM + col] =  ck[i] + vbk;
        v[row * DIM + col] =  cv[i] + vbv;
    }
}

// ---------------------------------------------------------------------------
// Kernel 2: edge attention + scatter-add. 128 threads per edge (one per dim).
// Wave-uniform edge metadata forced through readfirstlane -> scalar loads.
// ---------------------------------------------------------------------------
__global__ __launch_bounds__(256) void edge_kernel(
        const int*   __restrict__ eids,
        const float* __restrict__ ew,
        const float* __restrict__ cxcy,
        const float* __restrict__ Wp, const float* __restrict__ bp,
        const float* __restrict__ q,  const float* __restrict__ k,
        const float* __restrict__ v,  float* __restrict__ agg) {
    const int t = blockIdx.x * blockDim.x + threadIdx.x;
    const int e = __builtin_amdgcn_readfirstlane(t >> 7);  // wave-uniform edge id
    if (e >= N_EDGES) return;
    const int j = t & 127;

    const int   src = eids[e];                 // scalar loads (SMEM path)
    const int   tgt = eids[N_EDGES + e];
    const float w   = ew[e];
    const float dx  = cxcy[2 * tgt]     - cxcy[2 * src];
    const float dy  = cxcy[2 * tgt + 1] - cxcy[2 * src + 1];

    const float pos = dx * Wp[2 * j] + dy * Wp[2 * j + 1] + bp[j];
    const float qv  = q[(size_t)src * DIM + j];            // pre-scaled by 0.25
    const float kv  = k[(size_t)tgt * DIM + j] + pos;
    float prod = qv * kv;
    #pragma unroll
    for (int m = 8; m >= 1; m >>= 1) prod += __shfl_xor(prod, m, 16);

    const float attn    = w * (1.0f / (1.0f + __expf(-prod)));
    const float contrib = v[(size_t)tgt * DIM + j] * attn;
    unsafeAtomicAdd(&agg[(size_t)src * DIM + j], contrib); // global_atomic_add_f32
}

// ---------------------------------------------------------------------------
// Kernel 3: out = in_feats + agg @ Wo^T, agg tile staged once in LDS.
// Block = 256 threads; wave w computes col tile w.
// ---------------------------------------------------------------------------
__global__ __launch_bounds__(256) void out_kernel(
        const float* __restrict__ agg,
        const float* __restrict__ Wo,
        const float* __restrict__ in_feats,
        float* __restrict__ out) {
    __shared__ float sa_t[16 * LDS_STRIDE];

    const int tid = threadIdx.x;
    const int tr  = blockIdx.x * 16;
    {
        const int r  = tid >> 4;
        const int c0 = (tid & 15) * 8;
        const float* ar = agg + (size_t)(tr + r) * DIM + c0;
        *(float4*)&sa_t[r * LDS_STRIDE + c0]     = *(const float4*)ar;
        *(float4*)&sa_t[r * LDS_STRIDE + c0 + 4] = *(const float4*)(ar + 4);
    }
    __syncthreads();

    const int lane = tid & 31;
    const int lo   = lane & 15;
    const int hi   = lane >> 4;
    const int tc   = (tid >> 5) * 16;

    const float* sa = &sa_t[lo * LDS_STRIDE + 2 * hi];
    const float* pw = Wo + (size_t)(tc + lo) * DIM + 2 * hi;

    v8f c = {};
    #pragma unroll
    for (int s = 0; s < 32; ++s) {
        v2f a  = *(const v2f*)(sa + 4 * s);
        v2f bb = *(const v2f*)(pw + 4 * s);
        c = __builtin_amdgcn_wmma_f32_16x16x4_f32(false, a, false, bb,
                                                  (short)0, c, false, false);
    }

    const int col = tc + lo;
    #pragma unroll
    for (int i = 0; i < 8; ++i) {
        const size_t row = (size_t)(tr + i + 8 * hi);
        out[row * DIM + col] = in_feats[row * DIM + col] + c[i];
    }
}

// ---------------------------------------------------------------------------
extern "C" void kernel_launch(void* const* d_in, const int* in_sizes, int n_in,
                              void* d_out, int out_size, void* d_ws, size_t ws_size,
                              hipStream_t stream) {
    const float* in_feats = (const float*)d_in[0];
    const int*   edge_ids = (const int*)  d_in[1];
    const float* ew       = (const float*)d_in[2];
    const float* cxcy     = (const float*)d_in[3];
    const float* gamma    = (const float*)d_in[4];
    const float* beta     = (const float*)d_in[5];
    const float* Wq       = (const float*)d_in[6];
    const float* bq       = (const float*)d_in[7];
    const float* Wk       = (const float*)d_in[8];
    const float* bk       = (const float*)d_in[9];
    const float* Wv       = (const float*)d_in[10];
    const float* bv       = (const float*)d_in[11];
    const float* Wp       = (const float*)d_in[12];
    const float* bp       = (const float*)d_in[13];
    const float* Wo       = (const float*)d_in[14];
    float* out = (float*)d_out;

    const size_t nd = (size_t)N_NODES * DIM * sizeof(float);
    char* ws = (char*)d_ws;
    float* q   = (float*)(ws + 0 * nd);
    float* k   = (float*)(ws + 1 * nd);
    float* v   = (float*)(ws + 2 * nd);
    float* agg = (float*)(ws + 3 * nd);

    // 1) fused LayerNorm + q/k/v (LDS-staged A, 768 WMMAs/block)
    lnqkv_kernel<<<N_NODES / 16, 256, 0, stream>>>(
        in_feats, gamma, beta, Wq, bq, Wk, bk, Wv, bv, q, k, v);

    // 2) zero agg, then edge attention + hardware f32 atomic scatter
    hipMemsetAsync(agg, 0, nd, stream);
    edge_kernel<<<(N_EDGES * 128) / 256, 256, 0, stream>>>(
        edge_ids, ew, cxcy, Wp, bp, q, k, v, agg);

    // 3) out-projection + skip connection (LDS-staged A)
    out_kernel<<<N_NODES / 16, 256, 0, stream>>>(agg, Wo, in_feats, out);
}